// fcn_29008209117227
// MI455X (gfx1250) — compile-verified
//
#include <hip/hip_runtime.h>

#define DIN 2312
#define H1  800
#define O2  10
#define BB  32
#define TT  300
#define TB  (TT*BB)              // 9600 rows in batched-time GEMMs
#define DM  0.9512294245007140f  // exp(-1/20)
#define DS  0.8187307530779818f  // exp(-1/5)
#define INV_TM 0.05f
#define INV_TS 0.2f

typedef float v2f __attribute__((ext_vector_type(2)));
typedef float v8f __attribute__((ext_vector_type(8)));

// ---------------------------------------------------------------------------
// K1: layer-1 filtered traces. One thread per (b,i); sequential over t.
// x[b,i,t] contiguous in t (each thread streams its own cachelines).
// Writes E1a=(vm-vs), E1b=(vm/tm - vs/ts) in [t*B+b][i] layout (GEMM A layout).
// Also block-reduces the total input-spike count.
// ---------------------------------------------------------------------------
__global__ void k1_l1_traces(const float* __restrict__ x,
                             float* __restrict__ E1a, float* __restrict__ E1b,
                             float* __restrict__ sum_sp) {
  int idx = blockIdx.x * blockDim.x + threadIdx.x;
  float s = 0.f;
  if (idx < BB * DIN) {
    int b = idx / DIN, i = idx - b * DIN;
    const float* xp = x + (size_t)(b * DIN + i) * TT;
    float vm = 0.f, vs = 0.f;
    for (int t = 0; t < TT; ++t) {
      float xv = xp[t];
      s += xv;
      vm = vm * DM + xv;
      vs = vs * DS + xv;
      size_t o = (size_t)(t * BB + b) * DIN + i;
      E1a[o] = vm - vs;
      E1b[o] = vm * INV_TM - vs * INV_TS;
    }
  }
  __shared__ float red[256];
  red[threadIdx.x] = s;
  __syncthreads();
  for (int st = 128; st > 0; st >>= 1) {
    if (threadIdx.x < st) red[threadIdx.x] += red[threadIdx.x + st];
    __syncthreads();
  }
  if (threadIdx.x == 0) atomicAdd(&sum_sp[0], red[0]);
}

// ---------------------------------------------------------------------------
// Pad W2 (10x800) to W2p (16x800) with zero rows so the GEMM needs no
// per-iteration EXEC guard (zero rows contribute zero).
// ---------------------------------------------------------------------------
__global__ void k_pad_w2(const float* __restrict__ W2, float* __restrict__ W2p) {
  int idx = blockIdx.x * blockDim.x + threadIdx.x;   // 16*800
  if (idx >= 16 * H1) return;
  W2p[idx] = (idx < O2 * H1) ? W2[idx] : 0.f;
}

// ---------------------------------------------------------------------------
// K2: fused dual f32 WMMA GEMM.
//   C1 = A1 * W^T,  C2 = A2 * W^T   (B[k][n] = W[n*K+k], all N cols valid)
// One wave owns one 16-row M tile x NSTRIP 16-col N tiles, for BOTH A
// streams: per K-step it loads 1 A1-frag + 1 A2-frag + NSTRIP B-frags and
// issues 2*NSTRIP V_WMMA_F32_16X16X4_F32 — B fragments are reused across the
// two A streams, A fragments across the N strip.
// Fragment layout per ISA: lanes 0-15 hold K={k0,k0+1}, lanes 16-31 hold
// K={k0+2,k0+3}; per-lane loads are contiguous float2 in K.
// ---------------------------------------------------------------------------
template <int NSTRIP>
__global__ void __launch_bounds__(256)
k2_wmma_gemm_dual(const float* __restrict__ A1, const float* __restrict__ A2,
                  const float* __restrict__ W,
                  float* __restrict__ C1, float* __restrict__ C2,
                  int M, int Ktot, int ntN) {
  int lane = threadIdx.x & 31;
  int wave = threadIdx.x >> 5;
  int half = lane >> 4;
  int r    = lane & 15;
  int nStrips  = ntN / NSTRIP;
  int numUnits = (M >> 4) * nStrips;
  int unit = blockIdx.x * 8 + wave;
  if (unit >= numUnits) return;
  int mt = unit / nStrips, st = unit - mt * nStrips;
  int m0 = mt << 4, n0 = st * (NSTRIP << 4);

  size_t aoff = (size_t)(m0 + r) * Ktot + 2 * half;
  const float* A1row = A1 + aoff;
  const float* A2row = A2 + aoff;
  const float* Wrow[NSTRIP];
#pragma unroll
  for (int j = 0; j < NSTRIP; ++j)
    Wrow[j] = W + (size_t)(n0 + (j << 4) + r) * Ktot + 2 * half;

  v8f acc1[NSTRIP], acc2[NSTRIP];
#pragma unroll
  for (int j = 0; j < NSTRIP; ++j) {
    acc1[j] = (v8f){0.f, 0.f, 0.f, 0.f, 0.f, 0.f, 0.f, 0.f};
    acc2[j] = (v8f){0.f, 0.f, 0.f, 0.f, 0.f, 0.f, 0.f, 0.f};
  }

#pragma unroll 4
  for (int k0 = 0; k0 < Ktot; k0 += 4) {
    float2 a1v = *(const float2*)(A1row + k0);
    float2 a2v = *(const float2*)(A2row + k0);
    v2f a1; a1.x = a1v.x; a1.y = a1v.y;
    v2f a2; a2.x = a2v.x; a2.y = a2v.y;
#pragma unroll
    for (int j = 0; j < NSTRIP; ++j) {
      float2 bv = *(const float2*)(Wrow[j] + k0);
      v2f bm; bm.x = bv.x; bm.y = bv.y;
      acc1[j] = __builtin_amdgcn_wmma_f32_16x16x4_f32(false, a1, false, bm,
                                                      (short)0, acc1[j], false, false);
      acc2[j] = __builtin_amdgcn_wmma_f32_16x16x4_f32(false, a2, false, bm,
                                                      (short)0, acc2[j], false, false);
    }
  }

  int ldc = ntN << 4;
#pragma unroll
  for (int j = 0; j < NSTRIP; ++j) {
    size_t base = (size_t)(m0 + 8 * half) * ldc + n0 + (j << 4) + r;
#pragma unroll
    for (int q = 0; q < 8; ++q) {
      C1[base + (size_t)q * ldc] = acc1[j][q];
      C2[base + (size_t)q * ldc] = acc2[j][q];
    }
  }
}

// ---------------------------------------------------------------------------
// K3: layer-1 first-crossing scan + layer-2 filtered traces.
// One thread per (b,h). u1 = sav*(um1-us1) = sav*U1; du/dt = -D1.
// ---------------------------------------------------------------------------
__global__ void k3_l1_scan(const float* __restrict__ U1, const float* __restrict__ D1,
                           float* __restrict__ F2a, float* __restrict__ F2b,
                           int* __restrict__ t1, float* __restrict__ h1_dtdu,
                           float* __restrict__ sum_sp) {
  int idx = blockIdx.x * blockDim.x + threadIdx.x;
  float cnt = 0.f;
  if (idx < BB * H1) {
    int b = idx / H1, h = idx - b * H1;
    float sav = 1.f, vm = 0.f, vs = 0.f, dtdu = 0.f;
    int ts = TT;
    for (int t = 0; t < TT; ++t) {
      size_t o = (size_t)(t * BB + b) * H1 + h;
      float u = U1[o];
      float spk = 0.f;
      if (sav > 0.5f && u >= 1.0f) {
        spk = 1.f; sav = 0.f; ts = t; cnt += 1.f;
        float dudt = -D1[o];
        dudt = (dudt >= 0.f) ? fmaxf(dudt, 1e-2f) : fminf(dudt, -1e-2f);
        dtdu = 1.f / dudt;
      }
      vm = vm * DM + spk;
      vs = vs * DS + spk;
      F2a[o] = vm - vs;
      F2b[o] = vm * INV_TM - vs * INV_TS;
    }
    t1[idx] = ts;
    h1_dtdu[idx] = dtdu;
  }
  __shared__ float red[256];
  red[threadIdx.x] = cnt;
  __syncthreads();
  for (int st = 128; st > 0; st >>= 1) {
    if (threadIdx.x < st) red[threadIdx.x] += red[threadIdx.x + st];
    __syncthreads();
  }
  if (threadIdx.x == 0) atomicAdd(&sum_sp[1], red[0]);
}

// ---------------------------------------------------------------------------
// K5: layer-2 scan + readouts. One thread per (b,o). U2/D2 have ld=16.
// ---------------------------------------------------------------------------
__global__ void k5_l2_scan(const float* __restrict__ U2, const float* __restrict__ D2,
                           float* __restrict__ out_t, float* __restrict__ out_u,
                           float* __restrict__ h2_dtdu, int* __restrict__ t2,
                           float* __restrict__ sum_sp) {
  int idx = blockIdx.x * blockDim.x + threadIdx.x;
  if (idx >= BB * O2) return;
  int b = idx / O2, o = idx - b * O2;
  float sav = 1.f, uo = 0.f, dtdu = 0.f;
  int ts = TT;
  for (int t = 0; t < TT; ++t) {
    size_t off = (size_t)(t * BB + b) * 16 + o;
    float u = U2[off];
    if (sav > 0.5f && u >= 1.0f) {
      sav = 0.f; ts = t; uo = u;
      float dudt = -D2[off];
      dudt = (dudt >= 0.f) ? fmaxf(dudt, 1e-2f) : fminf(dudt, -1e-2f);
      dtdu = 1.f / dudt;
      atomicAdd(&sum_sp[2], 1.f);
    }
  }
  out_t[idx] = (float)ts;   // 300 + spk*(t-300) == t if spiked, else 300
  out_u[idx] = uo;
  h2_dtdu[idx] = dtdu;
  t2[idx] = ts;
}

// ---------------------------------------------------------------------------
// K6/K7: each neuron spikes at most once, so the outer-product accumulators
// are row gathers of the trace at the spike time (zero rows if no spike).
// ---------------------------------------------------------------------------
__global__ void k6_gather_h1(const float* __restrict__ E1a, const float* __restrict__ E1b,
                             const int* __restrict__ t1,
                             float* __restrict__ h1_v, float* __restrict__ h1_dvdt) {
  int row = blockIdx.y;                          // b*H1 + h  (25600 rows)
  int i = blockIdx.x * blockDim.x + threadIdx.x;
  if (i >= DIN) return;
  int b = row / H1;
  int ts = t1[row];
  size_t oo = (size_t)row * DIN + i;
  if (ts < TT) {
    size_t eo = (size_t)(ts * BB + b) * DIN + i;
    h1_v[oo] = E1a[eo];
    h1_dvdt[oo] = E1b[eo];
  } else {
    h1_v[oo] = 0.f;
    h1_dvdt[oo] = 0.f;
  }
}

__global__ void k7_gather_h2(const float* __restrict__ F2a, const float* __restrict__ F2b,
                             const int* __restrict__ t2,
                             float* __restrict__ h2_v, float* __restrict__ h2_dvdt) {
  int row = blockIdx.x;                          // b*O2 + o (320 rows)
  int b = row / O2;
  int ts = t2[row];
  for (int h = threadIdx.x; h < H1; h += blockDim.x) {
    size_t oo = (size_t)row * H1 + h;
    if (ts < TT) {
      size_t fo = (size_t)(ts * BB + b) * H1 + h;
      h2_v[oo] = F2a[fo];
      h2_dvdt[oo] = F2b[fo];
    } else {
      h2_v[oo] = 0.f;
      h2_dvdt[oo] = 0.f;
    }
  }
}

__global__ void k0_init(float* __restrict__ sum_sp) {
  if (threadIdx.x < 3) sum_sp[threadIdx.x] = 0.f;
}

// ---------------------------------------------------------------------------
extern "C" void kernel_launch(void* const* d_in, const int* in_sizes, int n_in,
                              void* d_out, int out_size, void* d_ws, size_t ws_size,
                              hipStream_t stream) {
  const float* x  = (const float*)d_in[0];   // (32, 2312, 300)
  const float* W1 = (const float*)d_in[1];   // (800, 2312)
  const float* W2 = (const float*)d_in[2];   // (10, 800)

  // Output layout (flat, return order)
  float* out      = (float*)d_out;
  float* out_t    = out;                                   // 320
  float* out_u    = out + 320;                             // 320
  float* sum_sp   = out + 640;                             // 3
  float* h1_v     = out + 643;                             // 32*800*2312
  float* h2_v     = h1_v    + (size_t)BB * H1 * DIN;       // 32*10*800
  float* h1_dvdt  = h2_v    + (size_t)BB * O2 * H1;
  float* h2_dvdt  = h1_dvdt + (size_t)BB * H1 * DIN;
  float* h1_dtdu  = h2_dvdt + (size_t)BB * O2 * H1;        // 32*800
  float* h2_dtdu  = h1_dtdu + (size_t)BB * H1;             // 320

  // Workspace (~302 MB): traces + batched GEMM results + spike times + W2 pad
  float* ws  = (float*)d_ws;
  size_t nE  = (size_t)TB * DIN;   // 22,195,200
  size_t nU1 = (size_t)TB * H1;    //  7,680,000
  size_t nU2 = (size_t)TB * 16;    //    153,600
  float* E1a = ws;
  float* E1b = E1a + nE;
  float* U1  = E1b + nE;
  float* D1  = U1 + nU1;
  float* F2a = D1 + nU1;
  float* F2b = F2a + nU1;
  float* U2  = F2b + nU1;
  float* D2  = U2 + nU2;
  float* W2p = D2 + nU2;           // 16*800
  int*   t1  = (int*)(W2p + 16 * H1);
  int*   t2  = t1 + BB * H1;

  k0_init<<<1, 32, 0, stream>>>(sum_sp);
  k_pad_w2<<<(16 * H1 + 255) / 256, 256, 0, stream>>>(W2, W2p);

  k1_l1_traces<<<(BB * DIN + 255) / 256, 256, 0, stream>>>(x, E1a, E1b, sum_sp);

  // GEMM1 (fused U1/D1): M=9600, K=2312, N=800; 600 M-tiles x 25 N-strips
  int units1 = (TB / 16) * ((H1 / 16) / 2);   // 15000
  k2_wmma_gemm_dual<2><<<(units1 + 7) / 8, 256, 0, stream>>>(
      E1a, E1b, W1, U1, D1, TB, DIN, H1 / 16);

  k3_l1_scan<<<(BB * H1 + 255) / 256, 256, 0, stream>>>(U1, D1, F2a, F2b, t1,
                                                        h1_dtdu, sum_sp);

  // GEMM2 (fused U2/D2): M=9600, K=800, N=16 (rows 10-15 of W2p are zero)
  int units2 = (TB / 16) * 1;                 // 600
  k2_wmma_gemm_dual<1><<<(units2 + 7) / 8, 256, 0, stream>>>(
      F2a, F2b, W2p, U2, D2, TB, H1, 1);

  k5_l2_scan<<<(BB * O2 + 63) / 64, 64, 0, stream>>>(U2, D2, out_t, out_u,
                                                     h2_dtdu, t2, sum_sp);

  dim3 g6((DIN + 255) / 256, BB * H1);
  k6_gather_h1<<<g6, 256, 0, stream>>>(E1a, E1b, t1, h1_v, h1_dvdt);
  k7_gather_h2<<<BB * O2, 256, 0, stream>>>(F2a, F2b, t2, h2_v, h2_dvdt);
}